// AugmentedTripletLoss_9388798509062
// MI455X (gfx1250) — compile-verified
//
#include <hip/hip_runtime.h>
#include <hip/hip_bf16.h>
#include <math.h>

#define N_ROWS 8192
#define DIM 128
#define NCENT 16
#define MARGIN_F 1.0f
#define EPS_F 1e-12f

#define ROWS_PER_WG 128
#define COLS_PER_ITER 128
#define LDS_STRIDE 132   // pad so lane->bank map is conflict free (n*132 mod 64 = 4n)

typedef __attribute__((ext_vector_type(2))) float v2f;
typedef __attribute__((ext_vector_type(8))) float v8f;

// ---------------------------------------------------------------------------
// Kernel 1: per-row squared norms, min distance to L2-normalized centers,
// and zero the output accumulator.
// ---------------------------------------------------------------------------
__global__ __launch_bounds__(256)
void atl_prep_kernel(const float* __restrict__ x,
                     const float* __restrict__ center,
                     float* __restrict__ sq,
                     float* __restrict__ dcmin,
                     float* __restrict__ out)
{
    __shared__ float cn[NCENT * DIM];
    __shared__ float cinv[NCENT];
    const int tid = threadIdx.x;

    if (tid < NCENT) {
        float s = 0.f;
        for (int k = 0; k < DIM; ++k) {
            float v = center[tid * DIM + k];
            s += v * v;
        }
        cinv[tid] = 1.0f / sqrtf(s);
    }
    __syncthreads();
    for (int i = tid; i < NCENT * DIM; i += blockDim.x)
        cn[i] = center[i] * cinv[i / DIM];
    __syncthreads();

    const int row = blockIdx.x * blockDim.x + tid;
    const float4* xr = (const float4*)(x + (size_t)row * DIM);

    float s = 0.f;
    float dot[NCENT];
#pragma unroll
    for (int p = 0; p < NCENT; ++p) dot[p] = 0.f;

    for (int kc = 0; kc < DIM / 4; ++kc) {
        float4 xv = xr[kc];
        s += xv.x * xv.x + xv.y * xv.y + xv.z * xv.z + xv.w * xv.w;
#pragma unroll
        for (int p = 0; p < NCENT; ++p) {
            float4 cv = ((const float4*)(cn + p * DIM))[kc];
            dot[p] += xv.x * cv.x + xv.y * cv.y + xv.z * cv.z + xv.w * cv.w;
        }
    }
    sq[row] = s;

    float dmin = __builtin_inff();
#pragma unroll
    for (int p = 0; p < NCENT; ++p) {
        // ||x - c||^2 = ||x||^2 - 2 x.c + 1   (||c|| == 1)
        float d2 = s - 2.0f * dot[p] + 1.0f;
        dmin = fminf(dmin, sqrtf(fmaxf(d2, EPS_F)));
    }
    dcmin[row] = dmin;

    if (row == 0) out[0] = 0.f;
}

// ---------------------------------------------------------------------------
// Kernel 2: fused Gram (f32 WMMA 16x16x4) + batch-hard mining + loss.
// One wave owns a 16-row tile; workgroup = 8 waves = 128 rows; columns are
// swept in 128-wide LDS-staged blocks shared by all 8 waves.
// B fragments for a whole tile are preloaded into registers so the 32-WMMA
// burst never waits on LDS; two accumulation chains give the scheduler
// independent WMMA streams.
// ---------------------------------------------------------------------------
__global__ __launch_bounds__(256)
void atl_main_kernel(const float* __restrict__ x,
                     const int* __restrict__ targets,
                     const float* __restrict__ sq,
                     const float* __restrict__ dcmin,
                     float* __restrict__ out)
{
    __shared__ float colX[COLS_PER_ITER * LDS_STRIDE];
    __shared__ float sqc[COLS_PER_ITER];
    __shared__ int   tcs[COLS_PER_ITER];

    const int tid  = threadIdx.x;
    const int wave = tid >> 5;
    const int lane = tid & 31;
    const int half = lane >> 4;   // 0: K+{0,1} / rows 0-7 of C; 1: K+{2,3} / rows 8-15
    const int ln   = lane & 15;

    const int rowTileBase = blockIdx.x * ROWS_PER_WG + wave * 16;

    // --- Preload A fragments for this wave's 16 rows, all of K=128.
    // ISA 32-bit A 16x4 layout: lanes 0-15 hold (M=lane, K=k0..k0+1),
    // lanes 16-31 hold (M=lane-16, K=k0+2..k0+3).
    v2f afrag[32];
    {
        const float* ap = x + (size_t)(rowTileBase + ln) * DIM + half * 2;
#pragma unroll
        for (int ks = 0; ks < 32; ++ks)
            afrag[ks] = *(const v2f*)(ap + ks * 4);
    }

    // Per-lane row metadata: this lane's 8 C rows are half*8 .. half*8+7.
    float sqr[8];
    int   trw[8];
#pragma unroll
    for (int r = 0; r < 8; ++r) {
        const int row = rowTileBase + half * 8 + r;
        sqr[r] = sq[row];
        trw[r] = targets[row];
    }

    float posMax[8], negMin[8];
#pragma unroll
    for (int r = 0; r < 8; ++r) {
        posMax[r] = -__builtin_inff();
        negMin[r] =  __builtin_inff();
    }

    for (int cb = 0; cb < N_ROWS; cb += COLS_PER_ITER) {
        __syncthreads();
        // Cooperative stage of a 128-column x 128-K block of X into LDS.
        for (int idx = tid; idx < COLS_PER_ITER * (DIM / 4); idx += 256) {
            const int c  = idx >> 5;      // column within block
            const int kc = idx & 31;      // float4 chunk along K
            float4 v = *(const float4*)(x + (size_t)(cb + c) * DIM + kc * 4);
            float* dst = &colX[c * LDS_STRIDE + kc * 4];
            dst[0] = v.x; dst[1] = v.y; dst[2] = v.z; dst[3] = v.w;
        }
        if (tid < COLS_PER_ITER) {
            sqc[tid] = sq[cb + tid];
            tcs[tid] = targets[cb + tid];
        }
        __syncthreads();

        for (int ct = 0; ct < COLS_PER_ITER / 16; ++ct) {
            // B 4x16 layout mirrors A: lane (ln,half) holds
            // (K=k0+2*half.., N=ln) -> contiguous 8B LDS reads, conflict free.
            // Preload the whole tile's B fragments so the WMMA burst does not
            // stall on ds round trips.
            const float* bp = &colX[(ct * 16 + ln) * LDS_STRIDE + half * 2];
            v2f bfrag[32];
#pragma unroll
            for (int ks = 0; ks < 32; ++ks)
                bfrag[ks] = *(const v2f*)(bp + ks * 4);

            // Two independent accumulation chains (even/odd K-steps).
            v8f acc0 = {};
            v8f acc1 = {};
#pragma unroll
            for (int ks = 0; ks < 32; ks += 2) {
                acc0 = __builtin_amdgcn_wmma_f32_16x16x4_f32(
                        false, afrag[ks + 0], false, bfrag[ks + 0], (short)0,
                        acc0, false, false);
                acc1 = __builtin_amdgcn_wmma_f32_16x16x4_f32(
                        false, afrag[ks + 1], false, bfrag[ks + 1], (short)0,
                        acc1, false, false);
            }

            const float sc  = sqc[ct * 16 + ln];
            const int   tcl = tcs[ct * 16 + ln];
#pragma unroll
            for (int r = 0; r < 8; ++r) {
                float g  = acc0[r] + acc1[r];
                float d2 = sqr[r] + sc - 2.0f * g;
                float d  = sqrtf(fmaxf(d2, EPS_F));
                const bool same = (trw[r] == tcl);
                posMax[r] = same ? fmaxf(posMax[r], d) : posMax[r];
                negMin[r] = same ? negMin[r] : fminf(negMin[r], d);
            }
        }
    }

    // Reduce across the 16 lanes of each half-wave (xor masks 1..8 stay
    // within a 16-lane group; each group holds a distinct set of 8 rows).
#pragma unroll
    for (int m = 8; m >= 1; m >>= 1) {
#pragma unroll
        for (int r = 0; r < 8; ++r) {
            posMax[r] = fmaxf(posMax[r], __shfl_xor(posMax[r], m, 32));
            negMin[r] = fminf(negMin[r], __shfl_xor(negMin[r], m, 32));
        }
    }

    if (ln == 0) {
        float lsum = 0.f;
#pragma unroll
        for (int r = 0; r < 8; ++r) {
            const int row = rowTileBase + half * 8 + r;
            const float dap = posMax[r];
            float dan = (negMin[r] < __builtin_inff()) ? negMin[r]
                                                       : dap + MARGIN_F;
            dan = fminf(dan, dcmin[row]);
            lsum += fmaxf(dap - dan + MARGIN_F, 0.f);
        }
        atomicAdd(out, lsum * (1.0f / (float)N_ROWS));
    }
}

// ---------------------------------------------------------------------------
extern "C" void kernel_launch(void* const* d_in, const int* in_sizes, int n_in,
                              void* d_out, int out_size, void* d_ws, size_t ws_size,
                              hipStream_t stream)
{
    const float* x       = (const float*)d_in[0];  // [8192,128] f32
    const int*   targets = (const int*)d_in[1];    // [8192]
    const float* center  = (const float*)d_in[2];  // [16,128] f32
    float* out = (float*)d_out;

    float* sq    = (float*)d_ws;       // [8192]
    float* dcmin = sq + N_ROWS;        // [8192]

    atl_prep_kernel<<<N_ROWS / 256, 256, 0, stream>>>(x, center, sq, dcmin, out);
    atl_main_kernel<<<N_ROWS / ROWS_PER_WG, 256, 0, stream>>>(
        x, targets, sq, dcmin, out);
}